// Triaffine_61881888801297
// MI455X (gfx1250) — compile-verified
//
#include <hip/hip_runtime.h>

typedef __attribute__((ext_vector_type(2))) float v2f;
typedef __attribute__((ext_vector_type(8))) float v8f;

#define N_IN 500
#define P    100        // N_PROJ
#define PB   101        // P + 1 (bias-augmented)
#define PPAD 112        // PB padded to 7*16
#define BDIM 2
#define LDIM 320
#define NTOT (PPAD * PPAD)   // 12544 = 784 tiles of 16
#define NTILES (NTOT / 16)   // 784
#define SLOPE 0.1f

__device__ __forceinline__ float lrelu(float v) { return v > 0.f ? v : SLOPE * v; }

// ---------------------------------------------------------------------------
// Kernel 0: Wt[k, i, j] = W[0, i, k, j], zero-padded to i,j < PPAD
// ---------------------------------------------------------------------------
__global__ void k_build_wt(const float* __restrict__ W, float* __restrict__ Wt) {
    int idx = blockIdx.x * 256 + threadIdx.x;
    if (idx >= P * NTOT) return;
    int k = idx / NTOT;
    int r = idx - k * NTOT;
    int i = r / PPAD, j = r - i * PPAD;
    float v = 0.f;
    if (i < PB && j < PB) v = W[(i * P + k) * PB + j];
    Wt[idx] = v;
}

// ---------------------------------------------------------------------------
// Kernel 1: projections.  xb/yb get the appended 1.0 at col 100, zeros to 112.
// NOTE: faithful to the reference — pz uses Wz applied to y.
// ---------------------------------------------------------------------------
__global__ void k_proj(const float* __restrict__ x, const float* __restrict__ y,
                       const float* __restrict__ Wx, const float* __restrict__ bx,
                       const float* __restrict__ Wy, const float* __restrict__ by,
                       const float* __restrict__ Wz, const float* __restrict__ bz,
                       float* __restrict__ xb, float* __restrict__ yb,
                       float* __restrict__ pz) {
    int bl = blockIdx.x;           // 0 .. B*L-1
    int t  = threadIdx.x;          // 128 threads
    if (t >= PPAD) return;
    const float* xr = x + (size_t)bl * N_IN;
    const float* yr = y + (size_t)bl * N_IN;
    float ox = 0.f, oy = 0.f, oz = 0.f;
    if (t < P) {
        const float* wx = Wx + (size_t)t * N_IN;
        const float* wy = Wy + (size_t)t * N_IN;
        const float* wz = Wz + (size_t)t * N_IN;
        float sx = 0.f, sy = 0.f, sz = 0.f;
        for (int i = 0; i < N_IN; ++i) {
            float xv = xr[i], yv = yr[i];
            sx += xv * wx[i];
            sy += yv * wy[i];
            sz += yv * wz[i];
        }
        ox = lrelu(sx + bx[t]);
        oy = lrelu(sy + by[t]);
        oz = lrelu(sz + bz[t]);
    } else if (t == P) {
        ox = 1.f; oy = 1.f; oz = 0.f;   // bias-augment columns
    }
    xb[(size_t)bl * PPAD + t] = ox;
    yb[(size_t)bl * PPAD + t] = oy;
    pz[(size_t)bl * PPAD + t] = oz;
}

// ---------------------------------------------------------------------------
// Kernel 2: M_all[b, z, i*PPAD+j] = sum_k pz[b,z,k] * Wt[k,i,j]
// One 16x16 output tile per wave; V_WMMA_F32_16X16X4_F32, 25 k-steps.
// Fragment layouts per CDNA5 ISA 7.12.2 (32-bit A 16x4, 32-bit C/D 16x16).
// ---------------------------------------------------------------------------
__global__ __launch_bounds__(256) void k_stageA(const float* __restrict__ pz,
                                                const float* __restrict__ Wt,
                                                float* __restrict__ Mall) {
    int wave = threadIdx.x >> 5;
    int lane = threadIdx.x & 31;
    int lrow = lane & 15, lhi = lane >> 4;
    int job  = blockIdx.x * 8 + wave;        // 0 .. 2*20*784-1 (exact)
    int nt   = job % NTILES;
    int rest = job / NTILES;
    int zt   = rest % 20;
    int b    = rest / 20;

    const float* Arow = pz + ((size_t)(b * LDIM + zt * 16 + lrow)) * PPAD;
    const float* Bcol = Wt + nt * 16 + lrow;

    v8f c = {0.f, 0.f, 0.f, 0.f, 0.f, 0.f, 0.f, 0.f};
    for (int kb = 0; kb < P; kb += 4) {
        int k0 = kb + 2 * lhi;                       // even -> 8B aligned
        v2f a = *(const v2f*)(Arow + k0);
        v2f bfrag;
        bfrag.x = Bcol[(size_t)k0 * NTOT];
        bfrag.y = Bcol[(size_t)(k0 + 1) * NTOT];
        c = __builtin_amdgcn_wmma_f32_16x16x4_f32(false, a, false, bfrag,
                                                  (short)0, c, false, false);
    }
    float* D = Mall + ((size_t)(b * LDIM) + zt * 16) * NTOT + nt * 16 + lrow;
#pragma unroll
    for (int v = 0; v < 8; ++v)
        D[(size_t)(v + 8 * lhi) * NTOT] = c[v];
}

// ---------------------------------------------------------------------------
// Kernel 3: per (b,z):  S = xb[b] @ M_z @ yb[b]^T   (320 x 320 output)
//   hop1: t[i,n] = sum_j M_z[i,j] * yb[yt*16+n, j]   (waves 0..6, one i-tile each)
//   hop2: S[x,n] = sum_i xb[x,i] * t[i,n]            (x-tiles over all 8 waves)
// ---------------------------------------------------------------------------
__global__ __launch_bounds__(256) void k_fused(const float* __restrict__ xb,
                                               const float* __restrict__ yb,
                                               const float* __restrict__ Mall,
                                               float* __restrict__ out) {
    __shared__ float Ml[PPAD * 113];   // M_z, padded stride (bank-conflict free)
    __shared__ float ybs[16 * 113];    // y-tile, transposed access
    __shared__ float tl[PPAD * 16];    // t = M_z @ ybT_tile

    int bz   = blockIdx.x;             // 0 .. 639
    int b    = bz / LDIM;
    int tid  = threadIdx.x;
    int wave = tid >> 5, lane = tid & 31;
    int lrow = lane & 15, lhi = lane >> 4;

    const float* Msrc = Mall + (size_t)bz * NTOT;
    for (int idx = tid; idx < NTOT; idx += 256) {
        int i = idx / PPAD, j = idx - i * PPAD;
        Ml[i * 113 + j] = Msrc[idx];
    }
    const float* ybB  = yb + (size_t)b * LDIM * PPAD;
    const float* xbB  = xb + (size_t)b * LDIM * PPAD;
    float*       outB = out + (size_t)bz * LDIM * LDIM;

    for (int yt = 0; yt < 20; ++yt) {
        __syncthreads();               // prev hop2 done (t, ybs free); Ml ready
        for (int idx = tid; idx < 16 * PPAD; idx += 256) {
            int n = idx / PPAD, j = idx - n * PPAD;
            ybs[n * 113 + j] = ybB[(size_t)(yt * 16 + n) * PPAD + j];
        }
        __syncthreads();

        // hop1
        if (wave < 7) {
            int it = wave;
            v8f c = {0.f, 0.f, 0.f, 0.f, 0.f, 0.f, 0.f, 0.f};
            for (int jb = 0; jb < PPAD; jb += 4) {
                int j0 = jb + 2 * lhi;
                v2f a;
                a.x = Ml[(it * 16 + lrow) * 113 + j0];
                a.y = Ml[(it * 16 + lrow) * 113 + j0 + 1];
                v2f bfrag;
                bfrag.x = ybs[lrow * 113 + j0];
                bfrag.y = ybs[lrow * 113 + j0 + 1];
                c = __builtin_amdgcn_wmma_f32_16x16x4_f32(false, a, false, bfrag,
                                                          (short)0, c, false, false);
            }
#pragma unroll
            for (int v = 0; v < 8; ++v)
                tl[(it * 16 + v + 8 * lhi) * 16 + lrow] = c[v];
        }
        __syncthreads();

        // hop2
        for (int xt = wave; xt < 20; xt += 8) {
            v8f c = {0.f, 0.f, 0.f, 0.f, 0.f, 0.f, 0.f, 0.f};
            for (int ib = 0; ib < PPAD; ib += 4) {
                int i0 = ib + 2 * lhi;
                v2f a = *(const v2f*)(xbB + (size_t)(xt * 16 + lrow) * PPAD + i0);
                v2f bfrag;
                bfrag.x = tl[i0 * 16 + lrow];
                bfrag.y = tl[(i0 + 1) * 16 + lrow];
                c = __builtin_amdgcn_wmma_f32_16x16x4_f32(false, a, false, bfrag,
                                                          (short)0, c, false, false);
            }
#pragma unroll
            for (int v = 0; v < 8; ++v)
                outB[(size_t)(xt * 16 + v + 8 * lhi) * LDIM + yt * 16 + lrow] = c[v];
        }
    }
}

// ---------------------------------------------------------------------------
extern "C" void kernel_launch(void* const* d_in, const int* in_sizes, int n_in,
                              void* d_out, int out_size, void* d_ws, size_t ws_size,
                              hipStream_t stream) {
    (void)in_sizes; (void)n_in; (void)out_size; (void)ws_size;
    const float* x  = (const float*)d_in[0];
    const float* y  = (const float*)d_in[1];
    // d_in[2] = z : unused by the reference computation
    const float* Wx = (const float*)d_in[3];
    const float* bx = (const float*)d_in[4];
    const float* Wy = (const float*)d_in[5];
    const float* by = (const float*)d_in[6];
    const float* Wz = (const float*)d_in[7];
    const float* bz = (const float*)d_in[8];
    const float* W  = (const float*)d_in[9];

    float* ws   = (float*)d_ws;
    float* xb   = ws;                                        //  71,680 f
    float* yb   = xb + (size_t)BDIM * LDIM * PPAD;           //  71,680 f
    float* pz   = yb + (size_t)BDIM * LDIM * PPAD;           //  71,680 f
    float* Wt   = pz + (size_t)BDIM * LDIM * PPAD;           // 1,254,400 f
    float* Mall = Wt + (size_t)P * NTOT;                     // 8,028,160 f (~38 MB total)

    float* out = (float*)d_out;

    k_build_wt<<<(P * NTOT + 255) / 256, 256, 0, stream>>>(W, Wt);
    k_proj<<<BDIM * LDIM, 128, 0, stream>>>(x, y, Wx, bx, Wy, by, Wz, bz, xb, yb, pz);
    k_stageA<<<(BDIM * 20 * NTILES) / 8, 256, 0, stream>>>(pz, Wt, Mall);
    k_fused<<<BDIM * LDIM, 256, 0, stream>>>(xb, yb, Mall, out);
}